// MMCE_30485677867028
// MI455X (gfx1250) — compile-verified
//
#include <hip/hip_runtime.h>
#include <math.h>

#define N_ROWS    8192
#define N_CLASSES 1000
#define STRIPS    (N_ROWS / 16)      // 512 row strips of 16
#define TILES_J   (N_ROWS / 16)      // 512 column tiles of 16
#define JSPLIT    8                  // split j-range for occupancy
#define TPW       (TILES_J / JSPLIT) // 64 tiles per wave
#define NPARTIAL  (STRIPS * JSPLIT)  // 4096 per-wave partials

// exp(-0.5*d*d) == exp2(-(S*d)^2) with S = sqrt(0.5*log2(e))
#define SQRT_HALF_LOG2E (0.84932180028904776f)

typedef __attribute__((ext_vector_type(2))) float v2f;
typedef __attribute__((ext_vector_type(8))) float v8f;

// ---------------------------------------------------------------------------
// Kernel 1: one wave32 per row. Coalesced float4 loads (1000/4 = 250 chunks),
// per-lane running (max, first-index), then shuffle-tree wave reduction with
// min-index tie-break (matches jnp.argmax first-occurrence semantics).
// ---------------------------------------------------------------------------
__global__ __launch_bounds__(256) void mmce_rowstats(
    const float* __restrict__ probs, const int* __restrict__ targets,
    float* __restrict__ conf, float* __restrict__ r) {
  const int lane = threadIdx.x & 31;
  const int wave = threadIdx.x >> 5;
  const int row  = blockIdx.x * 8 + wave;

  const float4* rowp = (const float4*)(probs + (size_t)row * N_CLASSES);
  float bestV = -1.0f;  // probs are in [0,1)
  int   bestI = 0;
  for (int c4 = lane; c4 < N_CLASSES / 4; c4 += 32) {
    float4 v = rowp[c4];
    int base = c4 * 4;
    if (v.x > bestV) { bestV = v.x; bestI = base;     }
    if (v.y > bestV) { bestV = v.y; bestI = base + 1; }
    if (v.z > bestV) { bestV = v.z; bestI = base + 2; }
    if (v.w > bestV) { bestV = v.w; bestI = base + 3; }
  }
  for (int off = 16; off > 0; off >>= 1) {
    float ov = __shfl_down(bestV, off, 32);
    int   oi = __shfl_down(bestI, off, 32);
    if (ov > bestV || (ov == bestV && oi < bestI)) { bestV = ov; bestI = oi; }
  }
  if (lane == 0) {
    float acc = (bestI == targets[row]) ? 1.0f : 0.0f;
    conf[row] = bestV;
    r[row]    = acc - bestV;
  }
}

// ---------------------------------------------------------------------------
// Kernel 2: one wave per 16-row strip x 64 column tiles.
// d'_ij = S*c_i - S*c_j via V_WMMA_F32_16X16X4_F32 as a rank-2 product
// (S = sqrt(0.5*log2e) pre-folded into the WMMA inputs, so the exp argument
// is just -(d')^2 — a single pk_mul with source-neg per element pair):
//   A columns: K0 = S*c_left, K1 = 1, K2 = K3 = 0 (documented 16x4 f32 layout)
//   B rows:    K0 = 1, K1 = -S*c_right; -S*c_right is placed in BOTH plausible
//              K1 slots so the result is independent of the (undocumented)
//              B VGPR ordering — the zero A columns kill the extra copy.
// Per element: v_exp_f32 (TRANS pipe) weighted by r_i (per-slot fmac) and
// r_j (folded once per tile). Loop is exp-throughput bound: 1 TRANS/pair.
// C/D layout: VGPR p holds M=p (lanes 0-15) / M=p+8 (lanes 16-31), N=lane&15.
// ---------------------------------------------------------------------------
__global__ __launch_bounds__(256) void mmce_pairwise(
    const float* __restrict__ conf, const float* __restrict__ r,
    float* __restrict__ partial) {
  const int lane    = threadIdx.x & 31;
  const int wave    = threadIdx.x >> 5;
  const int strip   = blockIdx.x * 8 + wave;   // 0..511
  const int i0      = strip * 16;
  const int laneLow = lane & 15;
  const bool lowHalf = (lane < 16);

  // A matrix (loop-invariant): col0 = S*c_left, col1 = ones, col2/3 = 0
  const float clS = conf[i0 + laneLow] * SQRT_HALF_LOG2E;
  v2f a;
  a.x = lowHalf ? clS  : 0.0f;
  a.y = lowHalf ? 1.0f : 0.0f;

  // r_left value for each D VGPR slot of this lane (loop-invariant)
  const int mbase = lowHalf ? 0 : 8;
  float rl[8];
#pragma unroll
  for (int p = 0; p < 8; ++p) rl[p] = r[i0 + mbase + p];

  float acc = 0.0f;
  const int jt0 = blockIdx.y * TPW;
  for (int jt = jt0; jt < jt0 + TPW; ++jt) {
    const int j0 = jt * 16;
    const float cRS = conf[j0 + laneLow] * SQRT_HALF_LOG2E;
    const float rR  = r[j0 + laneLow];
    v2f b;
    b.x = lowHalf ? 1.0f : -cRS;  // K0 row = ones; alt-layout K1 slot = -S*cR
    b.y = lowHalf ? -cRS : 0.0f;  // doc-layout K1 slot = -S*cR
    v8f c = {};
    v8f d = __builtin_amdgcn_wmma_f32_16x16x4_f32(
        /*neg_a=*/false, a, /*neg_b=*/false, b,
        /*c_mod=*/(short)0, c, /*reuse_a=*/false, /*reuse_b=*/false);
    float s = 0.0f;
#pragma unroll
    for (int p = 0; p < 8; ++p) {
      float dd = d[p];
      float k  = __builtin_amdgcn_exp2f(-dd * dd);  // v_exp_f32, neg-modifier mul
      s = fmaf(k, rl[p], s);
    }
    acc = fmaf(s, rR, acc);  // fold r_right once per tile
  }

  for (int off = 16; off > 0; off >>= 1) acc += __shfl_down(acc, off, 32);
  if (lane == 0) {
    int wid = (blockIdx.y * gridDim.x + blockIdx.x) * 8 + wave;
    partial[wid] = acc;
  }
}

// ---------------------------------------------------------------------------
// Kernel 3: single block; fixed-order (deterministic) reduction of the 4096
// partials plus sum(r^2) diagonal correction, in double for safety.
// ---------------------------------------------------------------------------
__global__ __launch_bounds__(256) void mmce_finalize(
    const float* __restrict__ r, const float* __restrict__ partial,
    float* __restrict__ out) {
  __shared__ double sdata[256];
  const int t = threadIdx.x;
  double s = 0.0;
  for (int i = t; i < NPARTIAL; i += 256) s += (double)partial[i];
  double rsq = 0.0;
  for (int i = t; i < N_ROWS; i += 256) { double rv = r[i]; rsq += rv * rv; }
  sdata[t] = s - rsq;
  __syncthreads();
  for (int off = 128; off > 0; off >>= 1) {
    if (t < off) sdata[t] += sdata[t + off];
    __syncthreads();
  }
  if (t == 0)
    out[0] = (float)(sdata[0] / ((double)N_ROWS * (double)(N_ROWS - 1)));
}

// ---------------------------------------------------------------------------
extern "C" void kernel_launch(void* const* d_in, const int* in_sizes, int n_in,
                              void* d_out, int out_size, void* d_ws, size_t ws_size,
                              hipStream_t stream) {
  const float* probs   = (const float*)d_in[0];
  const int*   targets = (const int*)d_in[1];
  float*       out     = (float*)d_out;

  float* conf    = (float*)d_ws;          // [8192]
  float* r       = conf + N_ROWS;         // [8192]
  float* partial = r + N_ROWS;            // [4096]

  // Phase 1: row max/argmax -> conf, r   (8 waves/block, 1 row/wave)
  mmce_rowstats<<<dim3(N_ROWS / 8), 256, 0, stream>>>(probs, targets, conf, r);

  // Phase 2: WMMA-tiled pairwise RBF quadratic form -> per-wave partials
  mmce_pairwise<<<dim3(STRIPS / 8, JSPLIT), 256, 0, stream>>>(conf, r, partial);

  // Phase 3: deterministic final reduction -> scalar
  mmce_finalize<<<dim3(1), 256, 0, stream>>>(r, partial, out);
}